// HybridConv_65481071410362
// MI455X (gfx1250) — compile-verified
//
#include <hip/hip_runtime.h>

// ---------------------------------------------------------------------------
// Fused 1x1-conv -> sigmoid -> [Linear+ReLU]x2 -> Linear(2) -> mean(sigmoid)
// Pure streaming elementwise kernel: 8 B/elem HBM traffic, ~25 flops/elem.
// Memory-bound by ~3 orders of magnitude => optimize data movement only.
// CDNA5 path: double-buffered GLOBAL_LOAD_ASYNC_TO_LDS_B128 staging with
// per-wave ASYNCcnt ordering (no barriers), NT stores for the write stream.
// ---------------------------------------------------------------------------

typedef float v4f __attribute__((ext_vector_type(4)));
typedef int   v4i __attribute__((vector_size(16)));   // matches builtin param type

#define THREADS 256          // 8 wave32s per block
#define MAX_BLOCKS 4096

#define AS1 __attribute__((address_space(1)))
#define AS3 __attribute__((address_space(3)))

// ---- device-pass feature probes (surface in compile stderr) ----------------
#if defined(__AMDGCN__)
  #if __has_builtin(__builtin_amdgcn_global_load_async_to_lds_b128)
    #define HAVE_ASYNC_LDS 1
  #else
    #warning DEVICE PROBE: __builtin_amdgcn_global_load_async_to_lds_b128 MISSING
    #define HAVE_ASYNC_LDS 0
  #endif
#else
  #define HAVE_ASYNC_LDS 0
#endif

struct P {
    float cw, cb;     // conv scalar weight/bias (THRESHOLD==0 folded in)
    float a1, c1;     // w1, b1
    float a2, c2;     // w2, b2
    float u0, v0;     // w3[0], b3[0]
    float u1, v1;     // w3[1], b3[1]
};

__device__ __forceinline__ float sigm(float t) {
    return 1.0f / (1.0f + __expf(-t));       // v_exp_f32; precision ample here
}

__device__ __forceinline__ float eval1(float x, const P& p) {
    float act = sigm(fmaf(x, p.cw, p.cb));
    float h   = fmaxf(fmaf(act, p.a1, p.c1), 0.0f);
    h         = fmaxf(fmaf(h,   p.a2, p.c2), 0.0f);
    return 0.5f * (sigm(fmaf(h, p.u0, p.v0)) + sigm(fmaf(h, p.u1, p.v1)));
}

__device__ __forceinline__ v4f eval4(v4f x, const P& p) {
    v4f r;
    r.x = eval1(x.x, p); r.y = eval1(x.y, p);
    r.z = eval1(x.z, p); r.w = eval1(x.w, p);
    return r;
}

// ---------------------------------------------------------------------------
// Main vectorized kernel. Grid-stride over 256-wide float4 tiles.
// Async path: each wave stages its own lanes via async-to-LDS and consumes
// only what it staged itself => per-wave ASYNCcnt suffices, no barriers.
// Fallback: direct b128 loads + global_prefetch.
// ---------------------------------------------------------------------------
__global__ void __launch_bounds__(THREADS)
mlp_stream_v4(const v4f* __restrict__ in4, v4f* __restrict__ out4,
              int n4, int nTiles,
              const float* __restrict__ cw, const float* __restrict__ cb,
              const float* __restrict__ w1, const float* __restrict__ b1,
              const float* __restrict__ w2, const float* __restrict__ b2,
              const float* __restrict__ w3, const float* __restrict__ b3)
{
    const P p = { cw[0], cb[0], w1[0], b1[0], w2[0], b2[0],
                  w3[0], b3[0], w3[1], b3[1] };
    const int tid    = (int)threadIdx.x;
    const int stride = (int)gridDim.x;
    int t = (int)blockIdx.x;
    if (t >= nTiles) return;   // uniform per block; no barriers used

#if HAVE_ASYNC_LDS
    __shared__ v4f buf[2][THREADS];
    {   // prologue: stage first tile into buffer 0
        const int idx = t * THREADS + tid;
        if (idx < n4) {
            __builtin_amdgcn_global_load_async_to_lds_b128(
                (AS1 v4i*)(in4 + idx),
                (AS3 v4i*)&buf[0][tid], 0, 0);
        }
    }
    int stage = 0;
    for (; t < nTiles; t += stride) {
        const int  tn   = t + stride;
        const bool more = (tn < nTiles);          // uniform per block
        if (more) {
            const int idxn = tn * THREADS + tid;
            if (idxn < n4) {
                __builtin_amdgcn_global_load_async_to_lds_b128(
                    (AS1 v4i*)(in4 + idxn),
                    (AS3 v4i*)&buf[stage ^ 1][tid], 0, 0);
            }
        }
        // oldest outstanding async op is tile t; wait for it (keep 1 in flight)
        if (more) asm volatile("s_wait_asynccnt 0x1" ::: "memory");
        else      asm volatile("s_wait_asynccnt 0x0" ::: "memory");

        const int idx = t * THREADS + tid;
        if (idx < n4) {
            v4f r = eval4(buf[stage][tid], p);
            __builtin_nontemporal_store(r, &out4[idx]);
        }
        stage ^= 1;
    }
#else
    for (; t < nTiles; t += stride) {
        const int idx = t * THREADS + tid;
        const int pf  = (t + stride) * THREADS + tid;
        if (pf < n4) __builtin_prefetch(&in4[pf], 0, 0);   // global_prefetch_b8
        if (idx < n4) {
            v4f r = eval4(in4[idx], p);
            __builtin_nontemporal_store(r, &out4[idx]);
        }
    }
#endif
}

// Scalar tail for n % 4 (not hit for N = 2^25, but kept for generality).
__global__ void mlp_tail(const float* __restrict__ in, float* __restrict__ out,
                         int start, int n,
                         const float* __restrict__ cw, const float* __restrict__ cb,
                         const float* __restrict__ w1, const float* __restrict__ b1,
                         const float* __restrict__ w2, const float* __restrict__ b2,
                         const float* __restrict__ w3, const float* __restrict__ b3)
{
    const P p = { cw[0], cb[0], w1[0], b1[0], w2[0], b2[0],
                  w3[0], b3[0], w3[1], b3[1] };
    int i = start + (int)(blockIdx.x * blockDim.x + threadIdx.x);
    if (i < n) out[i] = eval1(in[i], p);
}

extern "C" void kernel_launch(void* const* d_in, const int* in_sizes, int n_in,
                              void* d_out, int out_size, void* d_ws, size_t ws_size,
                              hipStream_t stream) {
    (void)n_in; (void)out_size; (void)d_ws; (void)ws_size;
    const float* data  = (const float*)d_in[0];
    const float* convw = (const float*)d_in[1];
    const float* convb = (const float*)d_in[2];
    const float* w1    = (const float*)d_in[3];
    const float* b1    = (const float*)d_in[4];
    const float* w2    = (const float*)d_in[5];
    const float* b2    = (const float*)d_in[6];
    const float* w3    = (const float*)d_in[7];
    const float* b3    = (const float*)d_in[8];
    float* out = (float*)d_out;

    const int n   = in_sizes[0];
    const int n4  = n >> 2;
    const int rem = n & 3;

    if (n4 > 0) {
        const int nTiles = (n4 + THREADS - 1) / THREADS;
        const int blocks = nTiles < MAX_BLOCKS ? nTiles : MAX_BLOCKS;
        mlp_stream_v4<<<blocks, THREADS, 0, stream>>>(
            (const v4f*)data, (v4f*)out, n4, nTiles,
            convw, convb, w1, b1, w2, b2, w3, b3);
    }
    if (rem > 0) {
        mlp_tail<<<1, 64, 0, stream>>>(data, out, n4 * 4, n,
                                       convw, convb, w1, b1, w2, b2, w3, b3);
    }
}